// GraphVAE_33715493273730
// MI455X (gfx1250) — compile-verified
//
#include <hip/hip_runtime.h>
#include <hip/hip_bf16.h>

typedef float v2f __attribute__((ext_vector_type(2)));
typedef float v8f __attribute__((ext_vector_type(8)));

#define F_DIM 128

// ---------------- degree / normalization ----------------

__global__ void deg_init(float* deg, int n) {
    int i = blockIdx.x * blockDim.x + threadIdx.x;
    if (i < n) deg[i] = 1.0f;  // self loop
}

__global__ void deg_count(float* deg, const int* __restrict__ dst, int e) {
    int i = blockIdx.x * blockDim.x + threadIdx.x;
    if (i < e) unsafeAtomicAdd(&deg[dst[i]], 1.0f);
}

__global__ void deg_finalize(float* deg, int n) {
    int i = blockIdx.x * blockDim.x + threadIdx.x;
    if (i < n) {
        float d = deg[i];
        deg[i] = (d > 0.0f) ? __frsqrt_rn(d) : 0.0f;  // dinv
    }
}

// ---------------- GEMM: H = X @ W  (N x 128 @ 128 x 128), f32 WMMA ----------------
// Block = 256 threads = 8 waves; block b handles rows [16b,16b+16); wave w handles cols [16w,16w+16).
// A 16x4 f32 layout: lanes 0-15 row=lane K={0,1}; lanes 16-31 row=lane-16 K={2,3}  -> contiguous float2.
// B 4x16 f32 layout: symmetric (row K striped across lanes).
// C/D: VGPR i -> row m0 + i + 8*(lane>>4), col n0 + (lane&15).

__global__ __launch_bounds__(256) void gemm_xw(const float* __restrict__ X,
                                               const float* __restrict__ W,
                                               float* __restrict__ H, int n) {
    const int m0 = blockIdx.x * 16;
    if (m0 >= n) return;
    const int lane = threadIdx.x & 31;
    const int wave = threadIdx.x >> 5;
    const int n0 = wave * 16;
    const int row = lane & 15;
    const int khalf = (lane >> 4) << 1;          // 0 or 2
    const int col = n0 + row;                    // B/D column for this lane

    const float2* __restrict__ xr2 =
        (const float2*)(X + (size_t)(m0 + row) * F_DIM);

    v8f acc = {};
#pragma unroll 8
    for (int k0 = 0; k0 < F_DIM; k0 += 4) {
        const int k = k0 + khalf;                // even
        float2 av = xr2[k >> 1];
        v2f a; a.x = av.x; a.y = av.y;
        v2f b;
        b.x = W[(size_t)k * F_DIM + col];
        b.y = W[(size_t)(k + 1) * F_DIM + col];
        acc = __builtin_amdgcn_wmma_f32_16x16x4_f32(
            /*neg_a=*/false, a, /*neg_b=*/false, b,
            /*c_mod=*/(short)0, acc, /*reuse_a=*/false, /*reuse_b=*/false);
    }

    const int rbase = m0 + ((lane >> 4) << 3);
    float* __restrict__ out = H + (size_t)rbase * F_DIM + col;
#pragma unroll
    for (int i = 0; i < 8; ++i)
        out[(size_t)i * F_DIM] = acc[i];
}

// ---------------- aggregation ----------------
// out[i] = bias + H[i] * dinv[i]^2   (self-loop + bias init; 1 wave per node, float4 lanes)
__global__ __launch_bounds__(256) void agg_init(const float* __restrict__ H,
                                                const float* __restrict__ dinv,
                                                const float* __restrict__ bias,
                                                float* __restrict__ out, int n) {
    int t = blockIdx.x * blockDim.x + threadIdx.x;
    int i = t >> 5;
    if (i >= n) return;
    int lane = t & 31;
    float di = dinv[i];
    float w = di * di;
    float4 v = ((const float4*)(H + (size_t)i * F_DIM))[lane];
    float4 bv = ((const float4*)bias)[lane];
    float4 o;
    o.x = bv.x + v.x * w; o.y = bv.y + v.y * w;
    o.z = bv.z + v.z * w; o.w = bv.w + v.w * w;
    ((float4*)(out + (size_t)i * F_DIM))[lane] = o;
}

// out[dst] += H[src] * dinv[src]*dinv[dst]   (1 wave per edge, fp32 global atomics)
__global__ __launch_bounds__(256) void agg_edges(const float* __restrict__ H,
                                                 const int* __restrict__ src,
                                                 const int* __restrict__ dst,
                                                 const float* __restrict__ dinv,
                                                 float* __restrict__ out, int e) {
    int t = blockIdx.x * blockDim.x + threadIdx.x;
    int ei = t >> 5;
    if (ei >= e) return;
    int lane = t & 31;
    int s = src[ei], d = dst[ei];
    float norm = dinv[s] * dinv[d];
    float4 v = ((const float4*)(H + (size_t)s * F_DIM))[lane];
    float* op = out + (size_t)d * F_DIM + lane * 4;
    unsafeAtomicAdd(op + 0, v.x * norm);
    unsafeAtomicAdd(op + 1, v.y * norm);
    unsafeAtomicAdd(op + 2, v.z * norm);
    unsafeAtomicAdd(op + 3, v.w * norm);
}

// ---------------- elementwise ----------------

__global__ void relu_inplace(float* x, int n4) {  // n4 = count of float4
    int i = blockIdx.x * blockDim.x + threadIdx.x;
    if (i >= n4) return;
    float4 v = ((float4*)x)[i];
    v.x = fmaxf(v.x, 0.0f); v.y = fmaxf(v.y, 0.0f);
    v.z = fmaxf(v.z, 0.0f); v.w = fmaxf(v.w, 0.0f);
    ((float4*)x)[i] = v;
}

__global__ void reparam(const float* __restrict__ eps,
                        const float* __restrict__ mu,
                        const float* __restrict__ lv,
                        float* __restrict__ z, int n) {
    int i = blockIdx.x * blockDim.x + threadIdx.x;
    if (i < n) z[i] = eps[i] * __expf(0.5f * lv[i]) + mu[i];
}

__global__ void sigmoid_inplace(float* x, int n) {
    int i = blockIdx.x * blockDim.x + threadIdx.x;
    if (i < n) x[i] = 1.0f / (1.0f + __expf(-x[i]));
}

// ---------------- driver ----------------

static inline int cdiv(int a, int b) { return (a + b - 1) / b; }

extern "C" void kernel_launch(void* const* d_in, const int* in_sizes, int n_in,
                              void* d_out, int out_size, void* d_ws, size_t ws_size,
                              hipStream_t stream) {
    const float* x   = (const float*)d_in[0];
    const int*   ei  = (const int*)d_in[1];
    const float* eps = (const float*)d_in[2];
    const float* W1  = (const float*)d_in[3];
    const float* b1  = (const float*)d_in[4];
    const float* Wmu = (const float*)d_in[5];
    const float* bmu = (const float*)d_in[6];
    const float* Wlv = (const float*)d_in[7];
    const float* blv = (const float*)d_in[8];

    const int N = in_sizes[0] / F_DIM;
    const int E = in_sizes[1] / 2;
    const int* src = ei;
    const int* dst = ei + E;

    // workspace layout
    float* dinv = (float*)d_ws;                 // N
    float* H    = dinv + N;                     // N*F (per-conv GEMM scratch)
    float* hbuf = H + (size_t)N * F_DIM;        // N*F (relu(conv1), later reused as z)

    float* recon = (float*)d_out;               // N*F
    float* mu    = recon + (size_t)N * F_DIM;   // N*F
    float* lv    = mu + (size_t)N * F_DIM;      // N*F

    const int TB = 256;
    const int nodeWaveBlocks = cdiv(N * 32, TB);
    const int edgeWaveBlocks = cdiv(E * 32, TB);
    const int gemmBlocks = cdiv(N, 16);
    const int nf = N * F_DIM;

    // degrees -> dinv
    deg_init<<<cdiv(N, TB), TB, 0, stream>>>(dinv, N);
    deg_count<<<cdiv(E, TB), TB, 0, stream>>>(dinv, dst, E);
    deg_finalize<<<cdiv(N, TB), TB, 0, stream>>>(dinv, N);

    // conv1: h = relu(gcn(x, W1, b1))   -> hbuf
    gemm_xw<<<gemmBlocks, TB, 0, stream>>>(x, W1, H, N);
    agg_init<<<nodeWaveBlocks, TB, 0, stream>>>(H, dinv, b1, hbuf, N);
    agg_edges<<<edgeWaveBlocks, TB, 0, stream>>>(H, src, dst, dinv, hbuf, E);
    relu_inplace<<<cdiv(nf / 4, TB), TB, 0, stream>>>(hbuf, nf / 4);

    // mu = gcn(h, Wmu, bmu)
    gemm_xw<<<gemmBlocks, TB, 0, stream>>>(hbuf, Wmu, H, N);
    agg_init<<<nodeWaveBlocks, TB, 0, stream>>>(H, dinv, bmu, mu, N);
    agg_edges<<<edgeWaveBlocks, TB, 0, stream>>>(H, src, dst, dinv, mu, E);

    // logvar = gcn(h, Wlv, blv)
    gemm_xw<<<gemmBlocks, TB, 0, stream>>>(hbuf, Wlv, H, N);
    agg_init<<<nodeWaveBlocks, TB, 0, stream>>>(H, dinv, blv, lv, N);
    agg_edges<<<edgeWaveBlocks, TB, 0, stream>>>(H, src, dst, dinv, lv, E);

    // z = eps * exp(0.5*logvar) + mu   (reuse hbuf)
    reparam<<<cdiv(nf, TB), TB, 0, stream>>>(eps, mu, lv, hbuf, nf);

    // recon = sigmoid(gcn(z, W1, b1))
    gemm_xw<<<gemmBlocks, TB, 0, stream>>>(hbuf, W1, H, N);
    agg_init<<<nodeWaveBlocks, TB, 0, stream>>>(H, dinv, b1, recon, N);
    agg_edges<<<edgeWaveBlocks, TB, 0, stream>>>(H, src, dst, dinv, recon, E);
    sigmoid_inplace<<<cdiv(nf, TB), TB, 0, stream>>>(recon, nf);
}